// LMEntity_70720931496065
// MI455X (gfx1250) — compile-verified
//
#include <hip/hip_runtime.h>
#include <hip/hip_bf16.h>

// H = 1024, E = 1024, V = 50257, M = 16384
#define HID 1024
#define VOCAB 50257
#define MEMSLOTS 16384

typedef __attribute__((ext_vector_type(2))) float v2f;
typedef __attribute__((ext_vector_type(8))) float v8f;

__device__ __forceinline__ float sigmoidf_(float x) { return 1.f / (1.f + expf(-x)); }

__device__ __forceinline__ float wave_sum(float v) {
  #pragma unroll
  for (int o = 16; o > 0; o >>= 1) v += __shfl_down(v, o, 32);
  return v;
}

// ---------------- K1: gates = W_ih @ emb[x] + W_hh @ h0 + b_ih + b_hh  (4096 rows)
__global__ void k_gates(const int* __restrict__ xp, const float* __restrict__ emb,
                        const float* __restrict__ Wih, const float* __restrict__ Whh,
                        const float* __restrict__ bih, const float* __restrict__ bhh,
                        const float* __restrict__ h0, float* __restrict__ gates) {
  int wave = blockIdx.x * 8 + (threadIdx.x >> 5);
  int lane = threadIdx.x & 31;
  const float* wr = Wih + (size_t)wave * HID;
  const float* vr = Whh + (size_t)wave * HID;
  const float* xe = emb + (size_t)xp[0] * HID;
  float acc = 0.f;
  #pragma unroll
  for (int i = 0; i < 8; ++i) {
    int idx = lane * 4 + i * 128;
    float4 a = *(const float4*)(wr + idx);
    float4 b = *(const float4*)(xe + idx);
    float4 c = *(const float4*)(vr + idx);
    float4 d = *(const float4*)(h0 + idx);
    acc += a.x * b.x + a.y * b.y + a.z * b.z + a.w * b.w;
    acc += c.x * d.x + c.y * d.y + c.z * d.z + c.w * d.w;
  }
  acc = wave_sum(acc);
  if (lane == 0) gates[wave] = acc + bih[wave] + bhh[wave];
}

// ---------------- K2: LSTM pointwise -> h  (gate order i,f,g,o)
__global__ void k_lstm(const float* __restrict__ gates, const float* __restrict__ c0,
                       float* __restrict__ h) {
  int j = blockIdx.x * 256 + threadIdx.x;
  float ig = sigmoidf_(gates[j]);
  float fg = sigmoidf_(gates[HID + j]);
  float gg = tanhf(gates[2 * HID + j]);
  float og = sigmoidf_(gates[3 * HID + j]);
  float c = fg * c0[j] + ig * gg;
  h[j] = og * tanhf(c);
}

// ---------------- K3: q partials: q = h @ Ws_w  (accumulate rows, 32 j-chunks)
__global__ void k_qpart(const float* __restrict__ h, const float* __restrict__ Ws,
                        float* __restrict__ qpart) {
  int t = threadIdx.x;
  int jb = blockIdx.x * 32;
  float a0 = 0.f, a1 = 0.f, a2 = 0.f, a3 = 0.f;
  for (int jj = 0; jj < 32; ++jj) {
    float hv = h[jb + jj];
    const float* row = Ws + (size_t)(jb + jj) * HID;
    a0 += hv * row[t];       a1 += hv * row[t + 256];
    a2 += hv * row[t + 512]; a3 += hv * row[t + 768];
  }
  float* qp = qpart + (size_t)blockIdx.x * HID;
  qp[t] = a0; qp[t + 256] = a1; qp[t + 512] = a2; qp[t + 768] = a3;
}

// ---------------- generic deterministic partial reduce (nparts x 1024 -> 1024)
__global__ void k_reduce(const float* __restrict__ part, float* __restrict__ out, int nparts) {
  int k = blockIdx.x * 256 + threadIdx.x;
  float s = 0.f;
  for (int b = 0; b < nparts; ++b) s += part[(size_t)b * HID + k];
  out[k] = s;
}

// ---------------- K5: scores[m] = q . h_e_m[m]  via V_WMMA_F32_16X16X4_F32
// Wave handles 16 rows. A: 16x4 f32 tile (lane=M, half-wave picks K pair).
// B: q replicated across all 16 columns -> every column of D equals scores.
__global__ void k_scores(const float* __restrict__ hem, const float* __restrict__ q,
                         float* __restrict__ scores) {
  __shared__ float qs[HID];
  for (int i = threadIdx.x; i < HID; i += 256) qs[i] = q[i];
  __syncthreads();
  int lane = threadIdx.x & 31;
  int wave = blockIdx.x * 8 + (threadIdx.x >> 5);
  int base = wave * 16;
  int m  = lane & 15;
  int kb = (lane >> 4) * 2;
  const float* arow = hem + (size_t)(base + m) * HID + kb;
  v8f acc0 = {0.f,0.f,0.f,0.f,0.f,0.f,0.f,0.f};
  v8f acc1 = {0.f,0.f,0.f,0.f,0.f,0.f,0.f,0.f};
  for (int k = 0; k < HID; k += 8) {
    v2f a0 = { arow[k],     arow[k + 1] };
    v2f b0 = { qs[k + kb],  qs[k + kb + 1] };
    v2f a1 = { arow[k + 4], arow[k + 5] };
    v2f b1 = { qs[k + 4 + kb], qs[k + 5 + kb] };
    acc0 = __builtin_amdgcn_wmma_f32_16x16x4_f32(false, a0, false, b0, (short)0, acc0, false, false);
    acc1 = __builtin_amdgcn_wmma_f32_16x16x4_f32(false, a1, false, b1, (short)0, acc1, false, false);
  }
  v8f acc = acc0 + acc1;
  if ((lane & 15) == 0) {           // N==0 lanes: lane0 -> rows 0..7, lane16 -> rows 8..15
    union { v8f v; float f[8]; } u; u.v = acc;
    int mb = (lane >> 4) * 8;
    #pragma unroll
    for (int vv = 0; vv < 8; ++vv) scores[base + mb + vv] = u.f[vv];
  }
}

// ---------------- K6: softmax over 16384 scores -> p_v (output)
__global__ void k_softmax(const float* __restrict__ scores, float* __restrict__ pv) {
  __shared__ float red[1024];
  int t = threadIdx.x;
  float loc[16];
  float mx = -3.4e38f;
  #pragma unroll
  for (int i = 0; i < 16; ++i) { loc[i] = scores[t + i * 1024]; mx = fmaxf(mx, loc[i]); }
  red[t] = mx; __syncthreads();
  for (int s = 512; s > 0; s >>= 1) { if (t < s) red[t] = fmaxf(red[t], red[t + s]); __syncthreads(); }
  mx = red[0]; __syncthreads();
  float sum = 0.f;
  #pragma unroll
  for (int i = 0; i < 16; ++i) { loc[i] = expf(loc[i] - mx); sum += loc[i]; }
  red[t] = sum; __syncthreads();
  for (int s = 512; s > 0; s >>= 1) { if (t < s) red[t] += red[t + s]; __syncthreads(); }
  float inv = 1.f / red[0];
  #pragma unroll
  for (int i = 0; i < 16; ++i) pv[t + i * 1024] = loc[i] * inv;
}

// ---------------- K7: d_i partials: d = p_v @ h_e_m (128 m-chunks)
__global__ void k_dpart(const float* __restrict__ hem, const float* __restrict__ pv,
                        float* __restrict__ dpart) {
  int t = threadIdx.x;
  int mb = blockIdx.x * 128;
  float a0 = 0.f, a1 = 0.f, a2 = 0.f, a3 = 0.f;
  for (int mm = 0; mm < 128; ++mm) {
    float p = pv[mb + mm];
    const float* row = hem + (size_t)(mb + mm) * HID;
    a0 += p * row[t];       a1 += p * row[t + 256];
    a2 += p * row[t + 512]; a3 += p * row[t + 768];
  }
  float* dp = dpart + (size_t)blockIdx.x * HID;
  dp[t] = a0; dp[t + 256] = a1; dp[t + 512] = a2; dp[t + 768] = a3;
}

// ---------------- K9: z = sigmoid(Wz . [h, d] + b)
__global__ void k_zgate(const float* __restrict__ h, const float* __restrict__ d,
                        const float* __restrict__ Wz, const float* __restrict__ Wzb,
                        float* __restrict__ zout) {
  __shared__ float red[256];
  int t = threadIdx.x;
  float s = 0.f;
  for (int j = t; j < HID; j += 256) s += Wz[j] * h[j] + Wz[HID + j] * d[j];
  red[t] = s; __syncthreads();
  for (int st = 128; st > 0; st >>= 1) { if (t < st) red[t] += red[t + st]; __syncthreads(); }
  if (t == 0) zout[0] = sigmoidf_(red[0] + Wzb[0]);
}

// ---------------- K10: u = tanh(W2 @ [h, h] + b)  (fold the two column halves)
__global__ void k_ugemv(const float* __restrict__ W2, const float* __restrict__ W2b,
                        const float* __restrict__ h, float* __restrict__ u) {
  int wave = blockIdx.x * 8 + (threadIdx.x >> 5);
  int lane = threadIdx.x & 31;
  const float* row = W2 + (size_t)wave * 2048;
  float acc = 0.f;
  #pragma unroll
  for (int i = 0; i < 8; ++i) {
    int idx = lane * 4 + i * 128;
    float4 a = *(const float4*)(row + idx);
    float4 b = *(const float4*)(row + 1024 + idx);
    float4 hv = *(const float4*)(h + idx);
    acc += (a.x + b.x) * hv.x + (a.y + b.y) * hv.y + (a.z + b.z) * hv.z + (a.w + b.w) * hv.w;
  }
  acc = wave_sum(acc);
  if (lane == 0) u[wave] = tanhf(acc + W2b[wave]);
}

// ---------------- K11: out = W1 @ u + b1  (50257 rows, WMMA tiles of 16)
__global__ void k_out(const float* __restrict__ W1, const float* __restrict__ W1b,
                      const float* __restrict__ u, float* __restrict__ out) {
  __shared__ float us[HID];
  for (int i = threadIdx.x; i < HID; i += 256) us[i] = u[i];
  __syncthreads();
  int lane = threadIdx.x & 31;
  int wave = blockIdx.x * 8 + (threadIdx.x >> 5);
  int base = wave * 16;
  if (base >= VOCAB) return;            // uniform per wave: EXEC stays all-ones
  int m  = lane & 15;
  int kb = (lane >> 4) * 2;
  int row = base + m; if (row > VOCAB - 1) row = VOCAB - 1;   // clamp tail loads
  const float* arow = W1 + (size_t)row * HID + kb;
  v8f acc0 = {0.f,0.f,0.f,0.f,0.f,0.f,0.f,0.f};
  v8f acc1 = {0.f,0.f,0.f,0.f,0.f,0.f,0.f,0.f};
  for (int k = 0; k < HID; k += 8) {
    v2f a0 = { arow[k],     arow[k + 1] };
    v2f b0 = { us[k + kb],  us[k + kb + 1] };
    v2f a1 = { arow[k + 4], arow[k + 5] };
    v2f b1 = { us[k + 4 + kb], us[k + 5 + kb] };
    acc0 = __builtin_amdgcn_wmma_f32_16x16x4_f32(false, a0, false, b0, (short)0, acc0, false, false);
    acc1 = __builtin_amdgcn_wmma_f32_16x16x4_f32(false, a1, false, b1, (short)0, acc1, false, false);
  }
  v8f acc = acc0 + acc1;
  if ((lane & 15) == 0) {
    union { v8f v; float f[8]; } uu; uu.v = acc;
    int mb = (lane >> 4) * 8;
    #pragma unroll
    for (int vv = 0; vv < 8; ++vv) {
      int r = base + mb + vv;
      if (r < VOCAB) out[r] = uu.f[vv] + W1b[r];
    }
  }
}

extern "C" void kernel_launch(void* const* d_in, const int* in_sizes, int n_in,
                              void* d_out, int out_size, void* d_ws, size_t ws_size,
                              hipStream_t stream) {
  const int*   x   = (const int*)  d_in[0];
  // d_in[1] entity_target: dead — new_mem is not returned and mem_read == h
  const float* emb = (const float*)d_in[2];
  const float* Wih = (const float*)d_in[3];
  const float* Whh = (const float*)d_in[4];
  const float* bih = (const float*)d_in[5];
  const float* bhh = (const float*)d_in[6];
  const float* h0  = (const float*)d_in[7];
  const float* c0  = (const float*)d_in[8];
  const float* hem = (const float*)d_in[9];
  const float* Wsw = (const float*)d_in[10];
  // d_in[11] Ws_b: adds a uniform shift to all scores -> cancelled by softmax
  const float* Wzw = (const float*)d_in[12];
  const float* Wzb = (const float*)d_in[13];
  const float* W2w = (const float*)d_in[14];
  const float* W2b = (const float*)d_in[15];
  const float* W1w = (const float*)d_in[16];
  const float* W1b = (const float*)d_in[17];

  float* ws     = (float*)d_ws;
  float* gates  = ws;                 // 4096
  float* h      = ws + 4096;          // 1024
  float* q      = ws + 5120;          // 1024
  float* scores = ws + 6144;          // 16384
  float* dvec   = ws + 22528;         // 1024
  float* uvec   = ws + 23552;         // 1024
  float* qpart  = ws + 24576;         // 32 * 1024
  float* dpart  = ws + 57344;         // 128 * 1024   (total ~736 KB)

  float* out  = (float*)d_out;        // 50257
  float* zout = out + VOCAB;          // 1
  float* pv   = out + VOCAB + 1;      // 16384

  k_gates  <<<512, 256, 0, stream>>>(x, emb, Wih, Whh, bih, bhh, h0, gates);
  k_lstm   <<<4,   256, 0, stream>>>(gates, c0, h);
  k_qpart  <<<32,  256, 0, stream>>>(h, Wsw, qpart);
  k_reduce <<<4,   256, 0, stream>>>(qpart, q, 32);
  k_scores <<<128, 256, 0, stream>>>(hem, q, scores);
  k_softmax<<<1,  1024, 0, stream>>>(scores, pv);
  k_dpart  <<<128, 256, 0, stream>>>(hem, pv, dpart);
  k_reduce <<<4,   256, 0, stream>>>(dpart, dvec, 128);
  k_zgate  <<<1,   256, 0, stream>>>(h, dvec, Wzw, Wzb, zout);
  k_ugemv  <<<128, 256, 0, stream>>>(W2w, W2b, h, uvec);
  k_out    <<<393, 256, 0, stream>>>(W1w, W1b, uvec, out);
}